// Avg2DPoolingMerger_82403242541301
// MI455X (gfx1250) — compile-verified
//
#include <hip/hip_runtime.h>
#include <hip/hip_bf16.h>
#include <cstdint>
#include <cstddef>

// Problem constants (match reference)
constexpr int Bn   = 64;
constexpr int Sn   = 1452;
constexpr int Dn   = 1024;
constexpr int Gn   = 288;
constexpr int Kn   = 4;
constexpr int Rn   = 32;
constexpr int VISn = 1152;
constexpr int MAXT = 300;
constexpr int OUTS = 600;   // MAXT + (Sn - VISn)

// ---------------------------------------------------------------------------
// Kernel 1: per-batch group validity, inverse counts, compaction order, and
// the full attention output row (written as float, per harness output dtype).
// One block per batch, Gn=288 threads (9 wave32s).
// ---------------------------------------------------------------------------
__global__ __launch_bounds__(Gn) void prep_kernel(
    const int* __restrict__ att_mask,   // (B,S) int32
    const int* __restrict__ pidx,       // (B,G,K) int32
    const int* __restrict__ rem,        // (B,R) int32
    int*       __restrict__ n_valid,    // (B,)
    int*       __restrict__ ord,        // (B,G) compacted valid group ids
    float*     __restrict__ inv_cnt,    // (B,G)
    float*     __restrict__ out_att)    // (B,OUTS) float
{
  const int b = blockIdx.x;
  const int g = threadIdx.x;            // 0..Gn-1

  __shared__ int s_rem[Rn];
  __shared__ int s_valid[Gn];
  __shared__ int s_nv;

  if (g < Rn) {
    int rv = rem[b * Rn + g];
    s_rem[g] = (rv == -1) ? -2 : rv;    // reference remaps -1 -> -2
  }
  __syncthreads();

  const int4 iv = *reinterpret_cast<const int4*>(pidx + ((size_t)b * Gn + g) * Kn);
  const int idxs[4] = {iv.x, iv.y, iv.z, iv.w};
  int cnt = 0;
#pragma unroll
  for (int k = 0; k < 4; ++k) {
    const int v = idxs[k];
    bool m = (v != -1);
    if (m) {
#pragma unroll
      for (int r = 0; r < Rn; ++r) m = m && (v != s_rem[r]);
    }
    cnt += m ? 1 : 0;
  }
  s_valid[g] = (cnt > 0) ? 1 : 0;
  inv_cnt[b * Gn + g] = 1.0f / (float)((cnt > 1) ? cnt : 1);
  __syncthreads();

  // Exclusive prefix sum over 288 flags (tiny; trivial per-thread scan).
  int rank = 0;
  for (int i = 0; i < g; ++i) rank += s_valid[i];
  if (s_valid[g]) ord[b * Gn + rank] = g;
  if (g == Gn - 1) {
    const int nv = rank + s_valid[g];
    n_valid[b] = nv;
    s_nv = nv;
  }
  __syncthreads();
  const int nv = s_nv;

  // Attention output: head = 1 on packed tail [MAXT-nv, MAXT), else 0;
  // tail = attention_mask[:, VIS:] (all written as float).
  for (int r = g; r < OUTS; r += Gn) {
    float a;
    if (r < MAXT) a = (r >= MAXT - nv) ? 1.0f : 0.0f;
    else          a = (float)att_mask[b * Sn + VISn + (r - MAXT)];
    out_att[(size_t)b * OUTS + r] = a;
  }
}

// ---------------------------------------------------------------------------
// Kernel 2: one block per output row (B*600 rows), 256 threads, float4/lane.
//  - r >= 300: tail copy hidden[b, VIS+(r-300), :] -> out row, through LDS
//              using gfx1250 async global<->LDS ops (ASYNCcnt path).
//  - r <  300: either zero row, or gather-4 + pooled write.
// Each output element is written exactly once.
// ---------------------------------------------------------------------------
__global__ __launch_bounds__(256) void merge_kernel(
    const float* __restrict__ hs,       // (B,S,D) f32
    const int*   __restrict__ prange,   // (B,2) int32
    const int*   __restrict__ pidx,     // (B,G,K) int32
    const int*   __restrict__ n_valid,  // (B,)
    const int*   __restrict__ ord,      // (B,G)
    const float* __restrict__ inv_cnt,  // (B,G)
    float*       __restrict__ out)      // (B,OUTS,D) f32
{
  __shared__ float4 buf[256];           // 4 KB staging for async tail copy

  const int blk = blockIdx.x;
  const int b   = blk / OUTS;
  const int r   = blk - b * OUTS;
  const int t   = threadIdx.x;

  float4* dst = reinterpret_cast<float4*>(out + ((size_t)b * OUTS + r) * Dn) + t;

  if (r >= MAXT) {
    // ---- tail: streaming copy via async tensor-data path (VGPR-free data) ----
    const float4* src =
        reinterpret_cast<const float4*>(hs + ((size_t)b * Sn + VISn + (r - MAXT)) * Dn) + t;
    // Low 32 bits of a flat-cast LDS pointer == byte offset in wave's LDS.
    unsigned lds_off = (unsigned)(uintptr_t)(&buf[t]);
    asm volatile("global_load_async_to_lds_b128 %0, %1, off"
                 :
                 : "v"(lds_off), "v"(src)
                 : "memory");
    asm volatile("s_wait_asynccnt 0x0" ::: "memory");
    asm volatile("global_store_async_from_lds_b128 %0, %1, off"
                 :
                 : "v"(dst), "v"(lds_off)
                 : "memory");
    // S_ENDPGM performs an implicit wait-idle, so the async store drains.
    return;
  }

  const int nv = n_valid[b];
  const int j  = r - (MAXT - nv);
  if (j < 0) {
    *dst = make_float4(0.f, 0.f, 0.f, 0.f);
    return;
  }

  const int g = ord[b * Gn + j];
  const int4 iv = *reinterpret_cast<const int4*>(pidx + ((size_t)b * Gn + g) * Kn);
  const int start = prange[b * 2 + 0];
  const int L     = prange[b * 2 + 1] - start + 1;

  const int r0 = start + (iv.x >= 0 ? iv.x : iv.x + L);
  const int r1 = start + (iv.y >= 0 ? iv.y : iv.y + L);
  const int r2 = start + (iv.z >= 0 ? iv.z : iv.z + L);
  const int r3 = start + (iv.w >= 0 ? iv.w : iv.w + L);

  const size_t base = (size_t)b * Sn;
  const float4 a0 = reinterpret_cast<const float4*>(hs + (base + r0) * Dn)[t];
  const float4 a1 = reinterpret_cast<const float4*>(hs + (base + r1) * Dn)[t];
  const float4 a2 = reinterpret_cast<const float4*>(hs + (base + r2) * Dn)[t];
  const float4 a3 = reinterpret_cast<const float4*>(hs + (base + r3) * Dn)[t];

  const float sc = inv_cnt[b * Gn + g];
  float4 o;
  o.x = (a0.x + a1.x + a2.x + a3.x) * sc;
  o.y = (a0.y + a1.y + a2.y + a3.y) * sc;
  o.z = (a0.z + a1.z + a2.z + a3.z) * sc;
  o.w = (a0.w + a1.w + a2.w + a3.w) * sc;
  *dst = o;
}

// ---------------------------------------------------------------------------
extern "C" void kernel_launch(void* const* d_in, const int* in_sizes, int n_in,
                              void* d_out, int out_size, void* d_ws, size_t ws_size,
                              hipStream_t stream) {
  (void)in_sizes; (void)n_in; (void)out_size; (void)ws_size;

  const float* hs     = (const float*)d_in[0];  // hidden_states (B,S,D) f32
  const int*   att    = (const int*)  d_in[1];  // attention_mask (B,S) i32
  const int*   prange = (const int*)  d_in[2];  // patch_range_list (B,2) i32
  const int*   pidx   = (const int*)  d_in[3];  // patch_indices (B,G,K) i32
  const int*   rem    = (const int*)  d_in[4];  // remove_index (B,R) i32

  float* out     = (float*)d_out;                       // (B,OUTS,D)
  float* out_att = out + (size_t)Bn * OUTS * Dn;        // (B,OUTS) as float

  // Workspace: n_valid[B] | ord[B*G] | inv_cnt[B*G]   (~150 KB)
  int*   nvalid = (int*)d_ws;
  int*   ordp   = nvalid + Bn;
  float* invc   = (float*)(ordp + (size_t)Bn * Gn);

  prep_kernel<<<Bn, Gn, 0, stream>>>(att, pidx, rem, nvalid, ordp, invc, out_att);
  merge_kernel<<<Bn * OUTS, 256, 0, stream>>>(hs, prange, pidx, nvalid, ordp, invc, out);
}